// TreeCRFSimple_18442589569874
// MI455X (gfx1250) — compile-verified
//
#include <hip/hip_runtime.h>
#include <math.h>

#define NN 128
#define LL 16

typedef float v2f __attribute__((ext_vector_type(2)));
typedef float v8f __attribute__((ext_vector_type(8)));

#define NEG_BIG (-3.0e38f)

// Packed upper-triangular index (j >= i), row-major by i.
__device__ __forceinline__ int triu(int i, int j) {
    return i * NN - ((i * (i - 1)) >> 1) + (j - i);
}

__global__ __launch_bounds__(256) void treecrf_inside_kernel(
    const float* __restrict__ lp,   // (B, N, N, L) fp32
    const int*   __restrict__ lens, // (B,) int32
    float*       __restrict__ out)  // (B,) fp32
{
    __shared__ float Alds[NN * NN];            // 64 KB: A[i][j] scalar DP table
    __shared__ float Plds[NN * (NN + 1) / 2];  // 33 KB: packed upper-tri P

    const int b    = blockIdx.x;
    const int tid  = threadIdx.x;
    const int lane = tid & 31;
    const int wave = tid >> 5;

    // ---------------- Stage 1: P[i][j] = logsumexp_l lp[b,i,j,:] ----------------
    const float* base = lp + (size_t)b * NN * NN * LL;
    for (int t = tid; t < NN * NN; t += 256) {
        int i = t >> 7;
        int j = t & (NN - 1);
        if (j < i) continue;  // lower triangle never used
        const float4* p = (const float4*)(base + (size_t)(i * NN + j) * LL);
        float4 x0 = p[0], x1 = p[1], x2 = p[2], x3 = p[3];
        float mx = fmaxf(fmaxf(fmaxf(x0.x, x0.y), fmaxf(x0.z, x0.w)),
                   fmaxf(fmaxf(fmaxf(x1.x, x1.y), fmaxf(x1.z, x1.w)),
                   fmaxf(fmaxf(fmaxf(x2.x, x2.y), fmaxf(x2.z, x2.w)),
                         fmaxf(fmaxf(x3.x, x3.y), fmaxf(x3.z, x3.w)))));
        float s = __expf(x0.x - mx) + __expf(x0.y - mx) + __expf(x0.z - mx) + __expf(x0.w - mx)
                + __expf(x1.x - mx) + __expf(x1.y - mx) + __expf(x1.z - mx) + __expf(x1.w - mx)
                + __expf(x2.x - mx) + __expf(x2.y - mx) + __expf(x2.z - mx) + __expf(x2.w - mx)
                + __expf(x3.x - mx) + __expf(x3.y - mx) + __expf(x3.z - mx) + __expf(x3.w - mx);
        float v = mx + __logf(s);
        Plds[triu(i, j)] = v;
        if (i == j) Alds[i * NN + i] = v;   // A diag init
    }
    __syncthreads();

    // ---------------- Stage 2: DP over diagonals -------------------------------
    // A[i,i+d] = P[i,i+d] + logsumexp_k( A[i,i+k] + A[i+k+1,i+d] ), k in [0,d).
    // Wave handles a 16-cell tile of the diagonal; cells <= 127 => <= 8 tiles.
    // WMMA f32 16x16x4 with B = ones acts as an exact 16-row x 4-term
    // multiply-accumulate engine for the shifted exponentials.
    v2f onesB; onesB.x = 1.0f; onesB.y = 1.0f;

    for (int d = 1; d < NN; ++d) {
        const int cells = NN - d;
        const int i0 = wave * 16;          // wave-uniform
        if (i0 < cells) {                  // wave-uniform branch: EXEC stays full
            const int row  = lane & 15;
            const int half = lane >> 4;
            int i = i0 + row; if (i > cells - 1) i = cells - 1;  // clamp dup rows
            const int j = i + d;
            const int rowBase = i * NN + i;   // &A[i][i]
            const int colBase = (i + 1) * NN + j; // &A[i+1][j]
            const int chunks = (d + 3) >> 2;

            // Pass 1: per-row max of (A[i,i+k] + A[i+k+1,j]) over k in [0,d)
            float m = NEG_BIG;
            for (int c = 0; c < chunks; ++c) {
                int k0 = 4 * c + 2 * half;
                int k1 = k0 + 1;
                int kk0 = (k0 < d) ? k0 : d - 1;
                int kk1 = (k1 < d) ? k1 : d - 1;
                float t0 = Alds[rowBase + kk0] + Alds[colBase + kk0 * NN];
                float t1 = Alds[rowBase + kk1] + Alds[colBase + kk1 * NN];
                t0 = (k0 < d) ? t0 : NEG_BIG;
                t1 = (k1 < d) ? t1 : NEG_BIG;
                m = fmaxf(m, fmaxf(t0, t1));
            }
            m = fmaxf(m, __shfl_xor(m, 16, 32));  // lanes L and L+16 share a row

            // Pass 2: WMMA row-sum of exp(t - m); B = ones so C[r][*] = sum_k.
            // Mask the *argument* (not the result): exp(NEG_BIG - m) == 0.0f,
            // keeping the inner loop branch-free (cndmask before the trans op).
            v8f acc = {};
            for (int c = 0; c < chunks; ++c) {
                int k0 = 4 * c + 2 * half;
                int k1 = k0 + 1;
                int kk0 = (k0 < d) ? k0 : d - 1;
                int kk1 = (k1 < d) ? k1 : d - 1;
                float t0 = Alds[rowBase + kk0] + Alds[colBase + kk0 * NN];
                float t1 = Alds[rowBase + kk1] + Alds[colBase + kk1 * NN];
                t0 = (k0 < d) ? t0 : NEG_BIG;
                t1 = (k1 < d) ? t1 : NEG_BIG;
                v2f a;
                a.x = __expf(t0 - m);   // == 0.0f for masked terms
                a.y = __expf(t1 - m);
                acc = __builtin_amdgcn_wmma_f32_16x16x4_f32(
                    /*neg_a=*/false, a, /*neg_b=*/false, onesB,
                    /*c_mod=*/(short)0, acc, /*reuse_a=*/false, /*reuse_b=*/false);
            }

            // Extract: VGPR r holds row r (lanes 0-15) / row 8+r (lanes 16-31).
            const int r = lane & 7;
            float s = acc[0];
            s = (r == 1) ? acc[1] : s;
            s = (r == 2) ? acc[2] : s;
            s = (r == 3) ? acc[3] : s;
            s = (r == 4) ? acc[4] : s;
            s = (r == 5) ? acc[5] : s;
            s = (r == 6) ? acc[6] : s;
            s = (r == 7) ? acc[7] : s;

            // Writer lanes: 0..7 -> rows 0..7 ; 16..23 -> rows 8..15.
            // Row (8+q)'s max lives in lanes (8+q)/(24+q): fetch via bpermute.
            float wm = __shfl(m, (lane >= 16) ? (lane - 8) : lane, 32);
            bool writer = (lane < 8) || (lane >= 16 && lane < 24);
            int wrow = (lane < 8) ? lane : (8 + (lane - 16));
            int wi = i0 + wrow;
            if (writer && wi < cells) {
                int wj = wi + d;
                Alds[wi * NN + wj] = Plds[triu(wi, wj)] + wm + __logf(s);
            }
        }
        __syncthreads();
    }

    // ---------------- Stage 3: output = A[0, len-1] ----------------------------
    if (tid == 0) {
        int len = lens[b];
        if (len < 1)  len = 1;
        if (len > NN) len = NN;
        out[b] = Alds[len - 1];   // row 0
    }
}

extern "C" void kernel_launch(void* const* d_in, const int* in_sizes, int n_in,
                              void* d_out, int out_size, void* d_ws, size_t ws_size,
                              hipStream_t stream) {
    (void)in_sizes; (void)n_in; (void)out_size; (void)d_ws; (void)ws_size;
    const float* lp   = (const float*)d_in[0];
    const int*   lens = (const int*)d_in[1];
    float*       out  = (float*)d_out;
    treecrf_inside_kernel<<<dim3(32), dim3(256), 0, stream>>>(lp, lens, out);
}